// SamplingBottleneckModule_27633819582789
// MI455X (gfx1250) — compile-verified
//
#include <hip/hip_runtime.h>
#include <hip/hip_bf16.h>
#include <stdint.h>

#define N_CLASSES   512
#define BLOCK       256
#define KTARGET     16.0f
#define LEVELS_M1   127.0f
#define INV_LEV_M1  (1.0f / 127.0f)

typedef __attribute__((ext_vector_type(4))) unsigned int tdm_v4u;
typedef __attribute__((ext_vector_type(8))) int          tdm_v8i;
typedef __attribute__((ext_vector_type(4))) int          tdm_v4i;

// ---- Tensor Data Mover: 1-D row (nelem fp32) global -> LDS -------------
__device__ __forceinline__ void tdm_load_row(uint32_t lds_off, const float* gsrc, int nelem) {
    unsigned long long ga = (unsigned long long)(uintptr_t)gsrc;
    tdm_v4u g0;
    g0.x = 1u;                                            // count=1 (valid), user mode
    g0.y = lds_off;                                       // lds_addr (bytes)
    g0.z = (unsigned)(ga & 0xFFFFFFFFull);                // global_addr[31:0]
    g0.w = (unsigned)((ga >> 32) & 0x01FFFFFFull)         // global_addr[56:32]
         | (2u << 30);                                    // type = 2 ("image")
    tdm_v8i g1;
    g1[0] = (2 << 16);                                    // data_size = 4B; wg_mask=0
    g1[1] = (int)((nelem & 0xFFFF) << 16);                // tensor_dim0[15:0] @bits63:48
    g1[2] = (int)(((nelem >> 16) & 0xFFFF) | (1 << 16));  // tensor_dim0 hi, tensor_dim1=1
    g1[3] = (int)((nelem & 0xFFFF) << 16);                // tensor_dim1 hi=0, tile_dim0
    g1[4] = 1;                                            // tile_dim1=1, tile_dim2=0
    g1[5] = nelem;                                        // tensor_dim0_stride lo
    g1[6] = 0;
    g1[7] = 0;
    tdm_v4i gz4 = {0, 0, 0, 0};                           // <=2D tensor: groups 2/3 zero
    tdm_v8i gz8 = {0, 0, 0, 0, 0, 0, 0, 0};               // extra words (clang-23 form)
    __builtin_amdgcn_tensor_load_to_lds(g0, g1, gz4, gz4, gz8, 0);
}

__device__ __forceinline__ uint32_t lds_byte_off(const void* p) {
    return (uint32_t)(uintptr_t)p;   // low 32 bits of flat addr = LDS offset
}

// ---- Block reductions (wave32 shuffle tree + per-stage LDS partials) ----
__device__ __forceinline__ float blockReduceMax(float a, float* buf, int tid) {
    #pragma unroll
    for (int o = 16; o > 0; o >>= 1) a = fmaxf(a, __shfl_xor(a, o, 32));
    if ((tid & 31) == 0) buf[tid >> 5] = a;
    __syncthreads();
    float r = buf[0];
    #pragma unroll
    for (int w = 1; w < 8; ++w) r = fmaxf(r, buf[w]);
    return r;
}

__device__ __forceinline__ float blockReduceSum(float a, float* buf, int tid) {
    #pragma unroll
    for (int o = 16; o > 0; o >>= 1) a += __shfl_xor(a, o, 32);
    if ((tid & 31) == 0) buf[tid >> 5] = a;
    __syncthreads();
    float r = 0.0f;
    #pragma unroll
    for (int w = 0; w < 8; ++w) r += buf[w];
    return r;
}

__device__ __forceinline__ float2 blockReduceSum2(float a, float b, float* buf, int tid) {
    #pragma unroll
    for (int o = 16; o > 0; o >>= 1) { a += __shfl_xor(a, o, 32); b += __shfl_xor(b, o, 32); }
    if ((tid & 31) == 0) { buf[tid >> 5] = a; buf[8 + (tid >> 5)] = b; }
    __syncthreads();
    float ra = 0.0f, rb = 0.0f;
    #pragma unroll
    for (int w = 0; w < 8; ++w) { ra += buf[w]; rb += buf[8 + w]; }
    return make_float2(ra, rb);
}

__global__ __launch_bounds__(BLOCK) void sampling_bottleneck_kernel(
    const float* __restrict__ logits, const float* __restrict__ values,
    const float* __restrict__ u_mask, const float* __restrict__ u_disc,
    float* __restrict__ out)
{
    __shared__ float s_vals[N_CLASSES];
    __shared__ float s_um[N_CLASSES];
    __shared__ float s_ud[N_CLASSES];
    __shared__ float s_red[6 * 16];   // one 16-float stage per reduction

    const int tid  = threadIdx.x;
    const int row  = blockIdx.x;
    const int base = row * N_CLASSES;

    // Kick off async TDM loads of the tail-phase inputs (wave 0 only; the
    // tensor op is wave-scalar). Overlaps DMA with softmax+Newton compute.
    if (tid < 32) {
        tdm_load_row(lds_byte_off(s_vals), values + base, N_CLASSES);
        tdm_load_row(lds_byte_off(s_um),   u_mask + base, N_CLASSES);
        tdm_load_row(lds_byte_off(s_ud),   u_disc + base, N_CLASSES);
    }

    // ---- softmax over the row (2 elements per thread, in registers) ----
    const float2 lg = reinterpret_cast<const float2*>(logits + base)[tid];
    const float mx  = blockReduceMax(fmaxf(lg.x, lg.y), s_red + 0 * 16, tid);

    const float e0 = __expf(lg.x - mx);
    const float e1 = __expf(lg.y - mx);
    const float s  = blockReduceSum(e0 + e1, s_red + 1 * 16, tid);
    const float inv_s = 1.0f / s;
    const float p0 = e0 * inv_s;
    const float p1 = e1 * inv_s;

    // ---- Newton solve: sum_i (1 - exp(-alpha p_i)) == K ----------------
    float alpha = KTARGET;
    #pragma unroll
    for (int it = 0; it < 3; ++it) {
        const float t0 = __expf(-alpha * p0);
        const float t1 = __expf(-alpha * p1);
        const float2 r = blockReduceSum2(t0 + t1, t0 * p0 + t1 * p1,
                                         s_red + (2 + it) * 16, tid);
        const float err = r.x + (KTARGET - (float)N_CLASSES);
        alpha += err / r.y;
    }

    // ---- marginals + straight-through discretization -------------------
    const float m0 = 1.0f - __expf(alpha * __logf(1.0f - p0));
    const float m1 = 1.0f - __expf(alpha * __logf(1.0f - p1));

    // Wait for the TDM DMAs, then make LDS visible to all waves.
    if (tid < 32) __builtin_amdgcn_s_wait_tensorcnt(0);
    __syncthreads();

    const float2 vv = reinterpret_cast<const float2*>(s_vals)[tid];
    const float2 um = reinterpret_cast<const float2*>(s_um)[tid];
    const float2 ud = reinterpret_cast<const float2*>(s_ud)[tid];

    const float y0 = floorf(m0 * LEVELS_M1 + ud.x) * INV_LEV_M1;
    const float y1 = floorf(m1 * LEVELS_M1 + ud.y) * INV_LEV_M1;

    float2 res;
    res.x = (um.x < y0) ? vv.x : 0.0f;
    res.y = (um.y < y1) ? vv.y : 0.0f;
    reinterpret_cast<float2*>(out + base)[tid] = res;
}

extern "C" void kernel_launch(void* const* d_in, const int* in_sizes, int n_in,
                              void* d_out, int out_size, void* d_ws, size_t ws_size,
                              hipStream_t stream) {
    const float* logits = (const float*)d_in[0];
    const float* values = (const float*)d_in[1];
    const float* u_mask = (const float*)d_in[2];
    const float* u_disc = (const float*)d_in[3];
    float* out = (float*)d_out;

    const int rows = in_sizes[0] / N_CLASSES;   // B*S = 16384 for the reference shapes
    sampling_bottleneck_kernel<<<dim3(rows), dim3(BLOCK), 0, stream>>>(
        logits, values, u_mask, u_disc, out);
}